// CalThroughput_50113678410435
// MI455X (gfx1250) — compile-verified
//
#include <hip/hip_runtime.h>
#include <hip/hip_bf16.h>

// ---------------------------------------------------------------------------
// MU-MIMO ZF throughput metric on MI455X (gfx1250).
// Heavy math (zero-padded 1024-pt DFT == [1024x32]x[32x32] complex GEMM) is
// done with full-precision V_WMMA_F32_16X16X4_F32 (f32 WMMA, matches the
// reference's complex64 arithmetic). Everything else is VALU f32.
// All reductions are fixed-order (no float atomics) => deterministic.
// ---------------------------------------------------------------------------

typedef float v2f __attribute__((ext_vector_type(2)));
typedef float v8f __attribute__((ext_vector_type(8)));

#define NCSUB 1024
#define NTANT 32
#define NGRP  256
#define NBAT  512
#define NTOT  (NGRP*NCSUB)      // 262144

// workspace layout (float offsets)
#define OFF_WR    0                         // [1024][32] cos(-2pi kq/1024)
#define OFF_WI    (OFF_WR + NCSUB*NTANT)    // [1024][32] -sin(...)
#define OFF_AR    (OFF_WI + NCSUB*NTANT)    // [32][32] cos(2pi np/32)/32
#define OFF_AI    (OFF_AR + NTANT*NTANT)    // [32][32] sin(2pi np/32)/32
#define OFF_TEMP  (OFF_AI + NTANT*NTANT)    // [1024 mats][2 comp][32 q][32 n]
#define OFF_HP    (OFF_TEMP + 1024*2*NTANT*NTANT)  // [N][8] unnormalized H'
#define OFF_PW    (OFF_HP + NTOT*8)         // [N][2] per-user gt power
#define OFF_PARTA (OFF_PW + NTOT*2)         // [256 g][8 chunk][4] {psum,pw0,pw1,pad}
#define OFF_PARTB (OFF_PARTA + NGRP*8*4)    // [256 g][4] {thr0,thr1,ideal0,ideal1}
// total ~ 4.80M floats ~ 19.2 MB

// ---------------------------------------------------------------------------
// Twiddle tables. Phase is reduced mod N as an integer before cosf/sinf so the
// argument stays in [0, 2pi) (full f32 accuracy).
// ---------------------------------------------------------------------------
__global__ void init_tables(float* ws) {
  int i = blockIdx.x * blockDim.x + threadIdx.x;
  const float TWO_PI = 6.28318530717958647692f;
  if (i < NCSUB * NTANT) {
    int k = i >> 5, q = i & 31;
    int ph = (k * q) & (NCSUB - 1);
    float th = TWO_PI * (float)ph * (1.0f / (float)NCSUB);
    ws[OFF_WR + i] =  cosf(th);     // Re e^{-j th}
    ws[OFF_WI + i] = -sinf(th);     // Im e^{-j th}
  } else {
    int j = i - NCSUB * NTANT;
    if (j < NTANT * NTANT) {
      int n = j >> 5, p = j & 31;
      int ph = (n * p) & 31;
      float th = TWO_PI * (float)ph * (1.0f / 32.0f);
      ws[OFF_AR + j] = cosf(th) * (1.0f / 32.0f);   // ifft: (1/N) e^{+j th}
      ws[OFF_AI + j] = sinf(th) * (1.0f / 32.0f);
    }
  }
}

// ---------------------------------------------------------------------------
// Antenna IDFT per 32x32 matrix: temp[n][q] = sum_p A32[n][p] * (c[p][q]-0.5-0.5j)
// Stored TRANSPOSED as [q][n] so it is directly the WMMA B operand (K=q, N=n).
// mi: 0..511 = pred batches, 512..1023 = gt batches.
// ---------------------------------------------------------------------------
__global__ void __launch_bounds__(128) temp_kernel(const float* __restrict__ gt,
                                                   const float* __restrict__ pred,
                                                   float* ws) {
  int mi  = blockIdx.x;
  int src = mi >> 9;              // 0 = pred, 1 = gt
  int b   = mi & 511;
  const float* in = (src ? gt : pred) + (size_t)b * 2 * NTANT * NTANT;

  __shared__ float cr[32][32], ci[32][32];
  int tid = threadIdx.x;
  for (int idx = tid; idx < 1024; idx += 128) {
    int p = idx >> 5, q = idx & 31;
    cr[p][q] = in[idx]        - 0.5f;
    ci[p][q] = in[1024 + idx] - 0.5f;
  }
  __syncthreads();

  const float* Ar = ws + OFF_AR;
  const float* Ai = ws + OFF_AI;
  float* tRe = ws + OFF_TEMP + mi * 2048;
  float* tIm = tRe + 1024;
  for (int o = tid; o < 1024; o += 128) {
    int q = o >> 5, n = o & 31;
    float sr = 0.f, si = 0.f;
    for (int p = 0; p < 32; ++p) {
      float ar = Ar[n * 32 + p], ai = Ai[n * 32 + p];
      float xr = cr[p][q],       xi = ci[p][q];
      sr += ar * xr - ai * xi;
      si += ar * xi + ai * xr;
    }
    tRe[o] = sr;   // [q][n]
    tIm[o] = si;
  }
}

// ---------------------------------------------------------------------------
// Main kernel: per (group, 128-subcarrier chunk).
//  Phase 1 (WMMA): sf[k][n] = sum_q W[k][q] * temp[n][q]  (complex, f32 WMMA)
//  Phase 2 (VALU): per-k ZF math reduced to 9 complex dots + 2x2 algebra.
// WMMA operand layouts per CDNA5 ISA 7.12.2:
//  A 16x4 f32 : lanes 0-15 M=lane hold K={0,1}; lanes 16-31 hold K={2,3}
//  B 4x16 f32 : lanes 0-15 N=lane hold K={0,1}; lanes 16-31 hold K={2,3}
//  C/D 16x16  : VGPR r -> M=r (lanes 0-15) / M=r+8 (lanes 16-31), N=lane%16
// f32 WMMA forbids A/B NEG bits, so the (-Wi*Ti) term folds the sign into B.
// ---------------------------------------------------------------------------
__global__ void __launch_bounds__(128) zf_kernel(float* ws) {
  int chunk = blockIdx.x;         // 0..7  -> k in [chunk*128, +128)
  int g     = blockIdx.y;         // 0..255
  int tid   = threadIdx.x, lane = tid & 31, wave = tid >> 5;

  __shared__ float tT[4][2][32][32];     // [m][re/im][q][n]  (m: 0,1=pred u0/u1; 2,3=gt)
  __shared__ float sfL[4][2][32][16];    // [m][re/im][antenna][k_local]  (k fastest: no bank conflicts)
  __shared__ float red[16][3];

  // stage temp planes for this group's 4 matrices
  float* tTf = &tT[0][0][0][0];
  for (int idx = tid; idx < 4 * 2048; idx += 128) {
    int m    = idx >> 11;
    int rest = idx & 2047;
    int src  = m >> 1;
    int b    = 2 * g + (m & 1);
    tTf[idx] = ws[OFF_TEMP + (src * 512 + b) * 2048 + rest];
  }
  __syncthreads();

  const float* Wr = ws + OFF_WR;
  const float* Wi = ws + OFF_WI;

  float psumAcc = 0.f, pwAcc0 = 0.f, pwAcc1 = 0.f;

  for (int kt = 0; kt < 8; ++kt) {
    int k0   = (chunk * 8 + kt) * 16;
    int krow = k0 + (lane & 15);
    int qoff = (lane >> 4) << 1;            // lanes>=16 carry K+2

    // each wave owns 4 of the 16 output tiles: (matrix, n-tile, re/im)
    for (int i = 0; i < 4; ++i) {
      int t    = wave * 4 + i;
      int m    = t >> 2;
      int nt   = (t >> 1) & 1;
      int part = t & 1;                     // 0 = Re, 1 = Im
      int nn   = nt * 16 + (lane & 15);
      v8f acc  = {0.f, 0.f, 0.f, 0.f, 0.f, 0.f, 0.f, 0.f};
      for (int ks = 0; ks < 8; ++ks) {
        int qa = ks * 4 + qoff;
        v2f aR, aI, bA, bB;
        aR.x = Wr[krow * 32 + qa];  aR.y = Wr[krow * 32 + qa + 1];
        aI.x = Wi[krow * 32 + qa];  aI.y = Wi[krow * 32 + qa + 1];
        float tr0 = tT[m][0][qa][nn], tr1 = tT[m][0][qa + 1][nn];
        float ti0 = tT[m][1][qa][nn], ti1 = tT[m][1][qa + 1][nn];
        if (part == 0) { bA.x = tr0; bA.y = tr1; bB.x = -ti0; bB.y = -ti1; }  // Re = Wr*Tr - Wi*Ti
        else           { bA.x = ti0; bA.y = ti1; bB.x =  tr0; bB.y =  tr1; }  // Im = Wr*Ti + Wi*Tr
        acc = __builtin_amdgcn_wmma_f32_16x16x4_f32(false, aR, false, bA, (short)0, acc, false, false);
        acc = __builtin_amdgcn_wmma_f32_16x16x4_f32(false, aI, false, bB, (short)0, acc, false, false);
      }
      int kb = (lane >> 4) << 3;            // lanes>=16 hold rows M=r+8
      for (int r = 0; r < 8; ++r) sfL[m][part][nn][kb + r] = acc[r];
    }
    __syncthreads();

    // ---- per-subcarrier ZF math (threads 0..15, one k each) ----
    if (tid < 16) {
      float g11 = 0, g22 = 0, g12r = 0, g12i = 0;
      float d10r = 0, d10i = 0, d11r = 0, d11i = 0;
      float d20r = 0, d20i = 0, d21r = 0, d21i = 0;
      float pw0 = 0, pw1 = 0;
      for (int a = 0; a < 32; ++a) {
        float a1r = sfL[0][0][a][tid], a1i = sfL[0][1][a][tid];
        float a2r = sfL[1][0][a][tid], a2i = sfL[1][1][a][tid];
        float h1r = sfL[2][0][a][tid], h1i = sfL[2][1][a][tid];
        float h2r = sfL[3][0][a][tid], h2i = sfL[3][1][a][tid];
        g11  += a1r * a1r + a1i * a1i;
        g22  += a2r * a2r + a2i * a2i;
        g12r += a1r * a2r + a1i * a2i;      // g12 = sum a1*conj(a2)
        g12i += a1i * a2r - a1r * a2i;
        d10r += h1r * a1r + h1i * a1i;  d10i += h1i * a1r - h1r * a1i;  // h1.conj(a1)
        d11r += h1r * a2r + h1i * a2i;  d11i += h1i * a2r - h1r * a2i;
        d20r += h2r * a1r + h2i * a1i;  d20i += h2i * a1r - h2r * a1i;
        d21r += h2r * a2r + h2i * a2i;  d21i += h2i * a2r - h2r * a2i;
        pw0  += h1r * h1r + h1i * h1i;
        pw1  += h2r * h2r + h2i * h2i;
      }
      // inv of hermitian 2x2 [[g11,g12],[g12*,g22]]
      float det = g11 * g22 - (g12r * g12r + g12i * g12i);
      float inv = 1.0f / det;
      float M00  = g22 * inv,  M11  = g11 * inv;        // real
      float M01r = -g12r * inv, M01i = -g12i * inv;
      float M10r = -g12r * inv, M10i =  g12i * inv;     // -conj(g12)/det
      // H'[u][v] = d_{u,0} M[0][v] + d_{u,1} M[1][v]   (unnormalized H = Hg P)
      float H00r = d10r * M00 + (d11r * M10r - d11i * M10i);
      float H00i = d10i * M00 + (d11r * M10i + d11i * M10r);
      float H01r = (d10r * M01r - d10i * M01i) + d11r * M11;
      float H01i = (d10r * M01i + d10i * M01r) + d11i * M11;
      float H10r = d20r * M00 + (d21r * M10r - d21i * M10i);
      float H10i = d20i * M00 + (d21r * M10i + d21i * M10r);
      float H11r = (d20r * M01r - d20i * M01i) + d21r * M11;
      float H11i = (d20r * M01i + d20i * M01r) + d21i * M11;
      // sum_t |P[t,0]+P[t,1]|^2 = |a|^2 g11 + |b|^2 g22 + 2Re(a*conj(b)*conj(g12))
      float alr = M00 + M01r, ali = M01i;
      float ber = M10r + M11, bei = M10i;
      float na = alr * alr + ali * ali;
      float nb = ber * ber + bei * bei;
      float acr = alr * ber + ali * bei;
      float aci = ali * ber - alr * bei;
      float psum = na * g11 + nb * g22 + 2.0f * (acr * g12r + aci * g12i);

      int n = g * 1024 + (k0 + tid);
      float* hp = ws + OFF_HP + n * 8;
      hp[0] = H00r; hp[1] = H00i; hp[2] = H01r; hp[3] = H01i;
      hp[4] = H10r; hp[5] = H10i; hp[6] = H11r; hp[7] = H11i;
      ws[OFF_PW + n * 2 + 0] = pw0;
      ws[OFF_PW + n * 2 + 1] = pw1;
      psumAcc += psum; pwAcc0 += pw0; pwAcc1 += pw1;
    }
    __syncthreads();
  }

  if (tid < 16) { red[tid][0] = psumAcc; red[tid][1] = pwAcc0; red[tid][2] = pwAcc1; }
  __syncthreads();
  if (tid == 0) {
    float s0 = 0, s1 = 0, s2 = 0;
    for (int i = 0; i < 16; ++i) { s0 += red[i][0]; s1 += red[i][1]; s2 += red[i][2]; }
    float* pa = ws + OFF_PARTA + (g * 8 + chunk) * 4;
    pa[0] = s0; pa[1] = s1; pa[2] = s2; pa[3] = 0.f;
  }
}

// ---------------------------------------------------------------------------
// Per-group: scale normalization + SINR/throughput sums over the 1024 k's.
// ---------------------------------------------------------------------------
__global__ void __launch_bounds__(128) final1_kernel(const float* __restrict__ snrp, float* ws) {
  int g = blockIdx.x, tid = threadIdx.x;
  float scaleSum = 0, pwSum0 = 0, pwSum1 = 0;
  for (int c = 0; c < 8; ++c) {
    const float* pa = ws + OFF_PARTA + (g * 8 + c) * 4;
    scaleSum += pa[0]; pwSum0 += pa[1]; pwSum1 += pa[2];
  }
  float scale = scaleSum * (1.0f / (1024.0f * 32.0f));
  float inv2  = 1.0f / (scale * scale);
  float lin   = powf(10.0f, snrp[0] * 0.1f);
  float s2_0  = sqrtf((pwSum0 * (1.0f / 1024.0f)) / lin);   // reference's sqrt quirk
  float s2_1  = sqrtf((pwSum1 * (1.0f / 1024.0f)) / lin);

  float thr0 = 0, thr1 = 0, id0 = 0, id1 = 0;
  for (int k = tid; k < 1024; k += 128) {
    int n = g * 1024 + k;
    const float* hp = ws + OFF_HP + n * 8;
    float a00 = (hp[0] * hp[0] + hp[1] * hp[1]) * inv2;
    float a01 = (hp[2] * hp[2] + hp[3] * hp[3]) * inv2;
    float a10 = (hp[4] * hp[4] + hp[5] * hp[5]) * inv2;
    float a11 = (hp[6] * hp[6] + hp[7] * hp[7]) * inv2;
    thr0 += log2f(1.0f + a00 / (a01 + s2_0));
    thr1 += log2f(1.0f + a11 / (a10 + s2_1));
    id0  += log2f(1.0f + ws[OFF_PW + n * 2 + 0] / s2_0);
    id1  += log2f(1.0f + ws[OFF_PW + n * 2 + 1] / s2_1);
  }
  __shared__ float red[128][4];
  red[tid][0] = thr0; red[tid][1] = thr1; red[tid][2] = id0; red[tid][3] = id1;
  __syncthreads();
  if (tid == 0) {
    float t0 = 0, t1 = 0, i0 = 0, i1 = 0;
    for (int i = 0; i < 128; ++i) { t0 += red[i][0]; t1 += red[i][1]; i0 += red[i][2]; i1 += red[i][3]; }
    float* pb = ws + OFF_PARTB + g * 4;
    pb[0] = t0; pb[1] = t1; pb[2] = i0; pb[3] = i1;
  }
}

__global__ void final2_kernel(const float* __restrict__ ws, float* __restrict__ out) {
  if (threadIdx.x == 0) {
    float t0 = 0, t1 = 0, i0 = 0, i1 = 0;
    for (int g = 0; g < NGRP; ++g) {
      const float* pb = ws + OFF_PARTB + g * 4;
      t0 += pb[0]; t1 += pb[1]; i0 += pb[2]; i1 += pb[3];
    }
    const float invN = 1.0f / (float)NTOT;
    t0 *= invN; t1 *= invN; i0 *= invN; i1 *= invN;
    out[0] = (i0 - t0) + (i1 - t1);   // distance_MU
    out[1] = t0 + t1;                 // throughput_MU
  }
}

extern "C" void kernel_launch(void* const* d_in, const int* in_sizes, int n_in,
                              void* d_out, int out_size, void* d_ws, size_t ws_size,
                              hipStream_t stream) {
  const float* gt   = (const float*)d_in[0];
  const float* pred = (const float*)d_in[1];
  const float* snr  = (const float*)d_in[2];
  float* ws  = (float*)d_ws;
  float* out = (float*)d_out;

  init_tables<<<dim3((NCSUB * NTANT + NTANT * NTANT + 255) / 256), 256, 0, stream>>>(ws);
  temp_kernel<<<dim3(1024), 128, 0, stream>>>(gt, pred, ws);
  zf_kernel<<<dim3(8, NGRP), 128, 0, stream>>>(ws);
  final1_kernel<<<dim3(NGRP), 128, 0, stream>>>(snr, ws);
  final2_kernel<<<dim3(1), 32, 0, stream>>>(ws, out);
}